// NMS_66657892434018
// MI455X (gfx1250) — compile-verified
//
#include <hip/hip_runtime.h>
#include <stdint.h>

constexpr int   NB     = 16;
constexpr int   NPT    = 25200;
constexpr int   CDIM   = 85;
constexpr int   KPRE   = 1024;
constexpr int   MAXDET = 1000;
constexpr float CONF_TH = 0.25f;
constexpr float IOU_TH  = 0.45f;
constexpr float MAXWH   = 4096.0f;
constexpr int   TILES_PB = 2080;   // 64*65/2 upper-triangle 16x16 tiles

typedef __attribute__((ext_vector_type(2))) float v2f;
typedef __attribute__((ext_vector_type(8))) float v8f;

__device__ __forceinline__ uint32_t f2ord(float f) {
    uint32_t b = __float_as_uint(f);
    return b ^ ((b >> 31) ? 0xFFFFFFFFu : 0x80000000u);
}
__device__ __forceinline__ float ord2f(uint32_t u) {
    uint32_t b = (u & 0x80000000u) ? (u ^ 0x80000000u) : (u ^ 0xFFFFFFFFu);
    return __uint_as_float(b);
}

// ---------------------------------------------------------------------------
// K1: per-candidate score/cls -> sortable key64 + cls.  64 rows per 256-thr
// block staged through LDS (coalesced); 4 threads per row scan 20 classes.
// ---------------------------------------------------------------------------
__global__ __launch_bounds__(256) void k_score(const float* __restrict__ x,
                                               unsigned long long* __restrict__ keys,
                                               uint32_t* __restrict__ clsA) {
    __shared__ float sb[64 * CDIM];
    __shared__ float pm[64][4];
    __shared__ int   pi[64][4];
    const int b    = blockIdx.y;
    const int row0 = blockIdx.x * 64;
    const int nrows = (NPT - row0 < 64) ? (NPT - row0) : 64;
    const float* src = x + ((size_t)b * NPT + row0) * CDIM;
    const int tot = nrows * CDIM;
    for (int i = threadIdx.x; i < tot; i += 256) sb[i] = src[i];
    __syncthreads();
    const int g = threadIdx.x >> 2, s = threadIdx.x & 3;
    if (g < nrows) {
        const float* p = &sb[g * CDIM];
        float m = -1e30f; int mi = 0;
        const int c0 = s * 20;
        for (int c = 0; c < 20; ++c) {
            float v = p[5 + c0 + c];
            if (v > m) { m = v; mi = c0 + c; }
        }
        pm[g][s] = m; pi[g][s] = mi;
    }
    __syncthreads();
    if (s == 0 && g < nrows) {
        float m = pm[g][0]; int mi = pi[g][0];
        for (int q = 1; q < 4; ++q)
            if (pm[g][q] > m) { m = pm[g][q]; mi = pi[g][q]; }
        const float obj  = sb[g * CDIM + 4];
        const float conf = obj * m;
        const bool  valid = (obj > CONF_TH) && (conf > CONF_TH);
        const float score = valid ? conf : -1.0f;
        const int n = row0 + g;
        keys[(size_t)b * NPT + n] =
            ((unsigned long long)f2ord(score) << 32) | (0xFFFFFFFFu - (uint32_t)n);
        clsA[(size_t)b * NPT + n] = (uint32_t)mi;
    }
}

// ---------------------------------------------------------------------------
// K2: bitonic top-1024 per batch (1 WG, 32 waves).  S kept sorted ascending;
// each chunk sorted descending, elementwise max, bitonic merge.  Emits
// selected fields (offset boxes, area, conf, cls) SoA to scratch.
// ---------------------------------------------------------------------------
__global__ __launch_bounds__(1024) void k_select(const float* __restrict__ x,
                                                 const unsigned long long* __restrict__ keys,
                                                 const uint32_t* __restrict__ clsA,
                                                 float* __restrict__ sel) {
    __shared__ unsigned long long S[KPRE];
    __shared__ unsigned long long Cb[KPRE];
    const int b = blockIdx.x, tid = threadIdx.x;
    const unsigned long long* kb = keys + (size_t)b * NPT;

    S[tid] = kb[tid];
    for (int k = 2; k <= KPRE; k <<= 1)
        for (int j = k >> 1; j > 0; j >>= 1) {
            __syncthreads();
            const int ixj = tid ^ j;
            if (ixj > tid) {
                const bool up = ((tid & k) == 0);
                unsigned long long a = S[tid], c = S[ixj];
                if ((a > c) == up) { S[tid] = c; S[ixj] = a; }
            }
        }
    const int nchunks = (NPT + KPRE - 1) / KPRE;
    for (int ch = 1; ch < nchunks; ++ch) {
        const int base = ch * KPRE;
        const int idx  = base + tid;
        Cb[tid] = (idx < NPT) ? kb[idx] : 0ull;
        if (base + KPRE + tid < NPT)
            __builtin_prefetch(&kb[base + KPRE + tid], 0, 3);   // global_prefetch_b8
        for (int k = 2; k <= KPRE; k <<= 1)                      // sort descending
            for (int j = k >> 1; j > 0; j >>= 1) {
                __syncthreads();
                const int ixj = tid ^ j;
                if (ixj > tid) {
                    const bool up = ((tid & k) != 0);
                    unsigned long long a = Cb[tid], c = Cb[ixj];
                    if ((a > c) == up) { Cb[tid] = c; Cb[ixj] = a; }
                }
            }
        __syncthreads();
        const unsigned long long m = (S[tid] > Cb[tid]) ? S[tid] : Cb[tid];
        S[tid] = m;                                              // bitonic; merge ascending
        for (int j = KPRE >> 1; j > 0; j >>= 1) {
            __syncthreads();
            const int ixj = tid ^ j;
            if (ixj > tid) {
                unsigned long long a = S[tid], c = S[ixj];
                if (a > c) { S[tid] = c; S[ixj] = a; }
            }
        }
    }
    __syncthreads();
    // rank r = tid (descending score), decode and emit fields
    const unsigned long long key = S[KPRE - 1 - tid];
    const uint32_t u   = (uint32_t)(key >> 32);
    const uint32_t idx = 0xFFFFFFFFu - (uint32_t)(key & 0xFFFFFFFFu);
    const float score  = ord2f(u);
    const float* p = x + ((size_t)b * NPT + idx) * CDIM;
    const float cx = p[0], cy = p[1], w = p[2], h = p[3];
    const float x1 = cx - 0.5f * w, y1 = cy - 0.5f * h;
    const float x2 = cx + 0.5f * w, y2 = cy + 0.5f * h;
    const float clsf = (float)clsA[(size_t)b * NPT + idx];
    const float off  = clsf * MAXWH;
    const size_t FS = (size_t)NB * KPRE;
    const size_t o  = (size_t)b * KPRE + tid;
    sel[0 * FS + o] = x1 + off;
    sel[1 * FS + o] = y1 + off;
    sel[2 * FS + o] = x2 + off;
    sel[3 * FS + o] = y2 + off;
    sel[4 * FS + o] = (x2 - x1) * (y2 - y1);
    sel[5 * FS + o] = score;
    sel[6 * FS + o] = clsf;
}

// ---------------------------------------------------------------------------
// K3: 16x16 IoU tiles, one wave per upper-triangle tile.  The area_i+area_j
// outer sum of each tile is a single V_WMMA_F32_16X16X4_F32 (A=[area_i,1],
// B=[1;area_j]); intersection max/min in VALU; bits packed via wave32 ballot.
// ---------------------------------------------------------------------------
__global__ __launch_bounds__(256) void k_iou(const float* __restrict__ sel,
                                             unsigned short* __restrict__ sup) {
    const int wave = blockIdx.x * 8 + (threadIdx.x >> 5);
    const int lane = threadIdx.x & 31;
    const int b = wave / TILES_PB;
    int rem = wave % TILES_PB;
    int ti = 0;
    while (rem >= (64 - ti)) { rem -= (64 - ti); ++ti; }
    const int tj = ti + rem;

    const size_t FS = (size_t)NB * KPRE;
    const float* ox1  = sel + 0 * FS + (size_t)b * KPRE;
    const float* oy1  = sel + 1 * FS + (size_t)b * KPRE;
    const float* ox2  = sel + 2 * FS + (size_t)b * KPRE;
    const float* oy2  = sel + 3 * FS + (size_t)b * KPRE;
    const float* area = sel + 4 * FS + (size_t)b * KPRE;

    const int  al = lane & 15;
    const bool hi = lane >= 16;
    const int  iBase = ti * 16, jBase = tj * 16;

    // A (16x4): lanes 0-15 hold K0=area_i[M], K1=1; lanes 16-31 (K2,K3)=0
    // B (4x16): lanes 0-15 hold K0=1, K1=area_j[N]; lanes 16-31 = 0
    v2f A;  A[0]  = hi ? 0.0f : area[iBase + al];  A[1]  = hi ? 0.0f : 1.0f;
    v2f Bm; Bm[0] = hi ? 0.0f : 1.0f;              Bm[1] = hi ? 0.0f : area[jBase + al];
    v8f acc = {0.f, 0.f, 0.f, 0.f, 0.f, 0.f, 0.f, 0.f};
#if __has_builtin(__builtin_amdgcn_wmma_f32_16x16x4_f32)
    v8f D = __builtin_amdgcn_wmma_f32_16x16x4_f32(false, A, false, Bm,
                                                  (short)0, acc, false, false);
#else
    v8f D;
    for (int rr = 0; rr < 8; ++rr)
        D[rr] = area[iBase + rr + (hi ? 8 : 0)] + area[jBase + al];
#endif

    const float jx1 = ox1[jBase + al], jy1 = oy1[jBase + al];
    const float jx2 = ox2[jBase + al], jy2 = oy2[jBase + al];

    const size_t rowBase = (size_t)b * KPRE * 64;
    for (int rr = 0; rr < 8; ++rr) {
        const int i = iBase + rr + (hi ? 8 : 0);
        const float ltx = fmaxf(ox1[i], jx1), lty = fmaxf(oy1[i], jy1);
        const float rbx = fminf(ox2[i], jx2), rby = fminf(oy2[i], jy2);
        const float iw = fmaxf(rbx - ltx, 0.0f), ih = fmaxf(rby - lty, 0.0f);
        const float inter = iw * ih;
        // iou > T  <=>  inter*(1+T) > T*(areasum + eps)   (denom > 0)
        const bool supb = inter * (1.0f + IOU_TH) > IOU_TH * (D[rr] + 1e-7f);
        const unsigned m = __builtin_amdgcn_ballot_w32(supb);
        if (lane == 0) {
            sup[rowBase + (size_t)(iBase + rr) * 64 + tj]     = (unsigned short)(m & 0xFFFFu);
            sup[rowBase + (size_t)(iBase + rr + 8) * 64 + tj] = (unsigned short)(m >> 16);
        }
    }
}

// ---------------------------------------------------------------------------
// K4: serial greedy suppression (matches fori_loop) with 256-row LDS chunks;
// wave0 walks the dependency chain wave-synchronously.  Then stable compact
// of kept entries (already sorted desc) into dets/vmask; rest zeroed.
// ---------------------------------------------------------------------------
__global__ __launch_bounds__(1024) void k_nms_out(const float* __restrict__ sel,
                                                  const unsigned int* __restrict__ supG,
                                                  float* __restrict__ dets,
                                                  float* __restrict__ vmask) {
    __shared__ unsigned int rowsL[256 * 32];
    __shared__ volatile unsigned int keepV[32];
    __shared__ unsigned int wsum[32];
    const int b = blockIdx.x, tid = threadIdx.x;
    const size_t FS = (size_t)NB * KPRE;
    const float* confA = sel + 5 * FS + (size_t)b * KPRE;

    {   // keep0 = conf > 0
        const bool k0 = confA[tid] > 0.0f;
        const unsigned m = __builtin_amdgcn_ballot_w32(k0);
        if ((tid & 31) == 0) keepV[tid >> 5] = m;
    }
    float* db = dets  + (size_t)b * MAXDET * 6;
    float* vb = vmask + (size_t)b * MAXDET;
    for (int i = tid; i < MAXDET * 6; i += 1024) db[i] = 0.0f;
    for (int i = tid; i < MAXDET;     i += 1024) vb[i] = 0.0f;
    __syncthreads();

    const unsigned int* supB = supG + (size_t)b * KPRE * 32;
    for (int chunk = 0; chunk < 4; ++chunk) {
        for (int k = tid; k < 256 * 32; k += 1024)
            rowsL[k] = supB[chunk * 256 * 32 + k];
        __syncthreads();
        if (tid < 32) {                         // single wave: lockstep serial chain
            const int t = tid;
            for (int ii = 0; ii < 256; ++ii) {
                const int i = chunk * 256 + ii;
                const unsigned ki = (keepV[i >> 5] >> (i & 31)) & 1u;
                if (ki) {
                    unsigned sw = rowsL[ii * 32 + t];
                    const int wi = i >> 5;
                    if (t < wi) sw = 0u;
                    else if (t == wi) sw &= ~((2u << (i & 31)) - 1u);  // mask j <= i
                    keepV[t] = keepV[t] & ~sw;
                }
            }
        }
        __syncthreads();
    }

    const unsigned myw = keepV[tid >> 5];
    const bool kept = (myw >> (tid & 31)) & 1u;
    if ((tid & 31) == 0) wsum[tid >> 5] = __popc(myw);
    __syncthreads();
    if (tid == 0) {
        unsigned s = 0;
        for (int w = 0; w < 32; ++w) { unsigned c = wsum[w]; wsum[w] = s; s += c; }
    }
    __syncthreads();
    if (kept) {
        const unsigned before = myw & ((1u << (tid & 31)) - 1u);
        const int pos = (int)wsum[tid >> 5] + __popc(before);
        if (pos < MAXDET) {
            const size_t o = (size_t)b * KPRE + tid;
            const float clsf = sel[6 * FS + o];
            const float off  = clsf * MAXWH;
            float* dr = db + (size_t)pos * 6;
            dr[0] = sel[0 * FS + o] - off;
            dr[1] = sel[1 * FS + o] - off;
            dr[2] = sel[2 * FS + o] - off;
            dr[3] = sel[3 * FS + o] - off;
            dr[4] = sel[5 * FS + o];
            dr[5] = clsf;
            vb[pos] = 1.0f;
        }
    }
}

// ---------------------------------------------------------------------------
extern "C" void kernel_launch(void* const* d_in, const int* in_sizes, int n_in,
                              void* d_out, int out_size, void* d_ws, size_t ws_size,
                              hipStream_t stream) {
    const float* x = (const float*)d_in[0];
    float* out = (float*)d_out;
    char* ws = (char*)d_ws;
    size_t off = 0;
    auto alloc = [&](size_t bytes) -> void* {
        off = (off + 255) & ~(size_t)255;
        void* p = ws + off;
        off += bytes;
        return p;
    };
    unsigned long long* keys = (unsigned long long*)alloc((size_t)NB * NPT * 8);
    uint32_t*           clsA = (uint32_t*)alloc((size_t)NB * NPT * 4);
    float*              sel  = (float*)alloc((size_t)7 * NB * KPRE * 4);
    unsigned short*     sup  = (unsigned short*)alloc((size_t)NB * KPRE * 64 * 2);

    dim3 g1((NPT + 63) / 64, NB);
    k_score<<<g1, 256, 0, stream>>>(x, keys, clsA);
    k_select<<<NB, 1024, 0, stream>>>(x, keys, clsA, sel);
    k_iou<<<(NB * TILES_PB) / 8, 256, 0, stream>>>(sel, sup);
    float* dets = out;
    float* vm   = out + (size_t)NB * MAXDET * 6;
    k_nms_out<<<NB, 1024, 0, stream>>>(sel, (const unsigned int*)sup, dets, vm);
}